// ImportanceWeightedMoE_31903017075151
// MI455X (gfx1250) — compile-verified
//
#include <hip/hip_runtime.h>
#include <math.h>

typedef __attribute__((ext_vector_type(2))) float v2f;
typedef __attribute__((ext_vector_type(8))) float v8f;

#define B_SZ 8192
#define H_SZ 1024
#define C_SZ 512
#define E_SZ 32
#define HID  4096   // 4*H
#define KIN  1536   // H + C
#define OUTD 1024

// workspace layout (float offsets)
#define WS_CAN    0                       // 32*512
#define WS_SORTW  (WS_CAN + 32*512)       // 8192*32
#define WS_COUNTS (WS_SORTW + 8192*32)    // 32
#define WS_IMP    (WS_COUNTS + 32)        // 32*1024
#define WS_IMPENT (WS_IMP + 32*1024)      // 32
#define WS_S      (WS_IMPENT + 32)        // 8192

__device__ __forceinline__ float wsum32(float v) {
#pragma unroll
  for (int m = 16; m > 0; m >>= 1) v += __shfl_xor(v, m, 32);
  return v;
}
__device__ __forceinline__ float wmax32(float v) {
#pragma unroll
  for (int m = 16; m > 0; m >>= 1) v = fmaxf(v, __shfl_xor(v, m, 32));
  return v;
}

// ---------------------------------------------------------------- init
__global__ void init_kernel(float* __restrict__ s, float* __restrict__ counts) {
  int i = blockIdx.x * 256 + threadIdx.x;
  if (i < B_SZ) s[i] = 0.f;
  if (i < E_SZ) counts[i] = 0.f;
}

// ---------------------------------------------------- normalize anchors
__global__ void norm_anchor_kernel(const float* __restrict__ ca, float* __restrict__ caN) {
  int e = blockIdx.x, lane = threadIdx.x;  // block = 32 (one wave)
  float v[16]; float n2 = 0.f;
#pragma unroll
  for (int j = 0; j < 16; ++j) { v[j] = ca[e * C_SZ + lane + 32 * j]; n2 = fmaf(v[j], v[j], n2); }
  n2 = wsum32(n2);
  float inv = 1.f / fmaxf(sqrtf(n2), 1e-8f);
#pragma unroll
  for (int j = 0; j < 16; ++j) caN[e * C_SZ + lane + 32 * j] = v[j] * inv;
}

// ------------------------------------------------------------- router
__global__ __launch_bounds__(256)
void router_kernel(const float* __restrict__ ce, const float* __restrict__ gum,
                   const float* __restrict__ caN, float* __restrict__ sortedW,
                   float* __restrict__ counts) {
  __shared__ float cPart[32];
  int tid = threadIdx.x, lane = tid & 31, wv = tid >> 5;
  if (tid < 32) cPart[tid] = 0.f;
  __syncthreads();

  int row = blockIdx.x * 8 + wv;
  float cer[16]; float n2 = 0.f;
#pragma unroll
  for (int j = 0; j < 16; ++j) { cer[j] = ce[row * C_SZ + lane + 32 * j]; n2 = fmaf(cer[j], cer[j], n2); }
  n2 = wsum32(n2);
  float inv = 1.f / fmaxf(sqrtf(n2), 1e-8f);

  float myLogit = 0.f;
  for (int e2 = 0; e2 < E_SZ; ++e2) {
    float p = 0.f;
#pragma unroll
    for (int j = 0; j < 16; ++j) p = fmaf(cer[j], caN[e2 * C_SZ + lane + 32 * j], p);
    p = wsum32(p);
    if (lane == e2) myLogit = p;
  }
  float z = (myLogit * inv * 0.125f + gum[row * E_SZ + lane]) * 10.0f;  // /tau
  float m = wmax32(z);
  float ex = expf(z - m);
  float w = ex / wsum32(ex);

  atomicAdd(&cPart[lane], w);
  // stable descending rank (ties -> smaller index first), matches lax.top_k
  int c = 0;
  for (int j = 0; j < 32; ++j) {
    float wj = __shfl(w, j, 32);
    c += (wj > w) || (wj == w && j < lane);
  }
  sortedW[row * E_SZ + c] = w;
  __syncthreads();
  if (tid < 32) atomicAdd(&counts[tid], cPart[tid]);
}

// --------------------------------------------------------- importance
__global__ __launch_bounds__(256)
void importance_kernel(const float* __restrict__ fi, const float* __restrict__ Tptr,
                       float* __restrict__ impOut, float* __restrict__ impEnt) {
  __shared__ float redM[8], redS[8], redE[8];
  int e = blockIdx.x, tid = threadIdx.x, lane = tid & 31, wv = tid >> 5;
  float temp = fminf(fmaxf(Tptr[0], 0.1f), 5.0f);
  float x[4];
#pragma unroll
  for (int j = 0; j < 4; ++j) x[j] = fi[e * H_SZ + tid + 256 * j] / temp;
  float m = fmaxf(fmaxf(x[0], x[1]), fmaxf(x[2], x[3]));
  m = wmax32(m);
  if (lane == 0) redM[wv] = m;
  __syncthreads();
  float M = redM[0];
#pragma unroll
  for (int i = 1; i < 8; ++i) M = fmaxf(M, redM[i]);
  float exs[4]; float ssum = 0.f;
#pragma unroll
  for (int j = 0; j < 4; ++j) { exs[j] = expf(x[j] - M); ssum += exs[j]; }
  ssum = wsum32(ssum);
  if (lane == 0) redS[wv] = ssum;
  __syncthreads();
  float S = 0.f;
#pragma unroll
  for (int i = 0; i < 8; ++i) S += redS[i];
  float ent = 0.f;
#pragma unroll
  for (int j = 0; j < 4; ++j) {
    float p = exs[j] / S;
    impOut[e * H_SZ + tid + 256 * j] = p;
    ent -= p * logf(p + 1e-8f);
  }
  ent = wsum32(ent);
  if (lane == 0) redE[wv] = ent;
  __syncthreads();
  if (tid == 0) { float Es = 0.f; for (int i = 0; i < 8; ++i) Es += redE[i]; impEnt[e] = Es; }
}

// ------------------------------------------------- fused MoE (WMMA f32)
__device__ __forceinline__ void epilogue_tile(const v8f& a, int ng, int e,
                                              const float* __restrict__ b1,
                                              const float* __restrict__ W2,
                                              float rowAcc[8]) {
  float bias = b1[e * HID + ng];
  float w2v  = W2[((size_t)e * HID + ng) * E_SZ + e];
#pragma unroll
  for (int r = 0; r < 8; ++r) {
    float xv = a[r] + bias;
    float g  = 0.5f * xv * (1.0f + erff(xv * 0.70710678118654752f));  // exact GELU
    rowAcc[r] = fmaf(g, w2v, rowAcc[r]);
  }
}

#define W1T_PITCH 34   // floats per n-row: 32 K values + 2 pad (even, b64-aligned, conflict-free loads)

__global__ __launch_bounds__(256)
void moe_kernel(const float* __restrict__ h, const float* __restrict__ code,
                const float* __restrict__ W1, const float* __restrict__ b1,
                const float* __restrict__ W2, const float* __restrict__ b2,
                const float* __restrict__ imp, const float* __restrict__ sortedW,
                float* __restrict__ s) {
  __shared__ __align__(16) float impLds[H_SZ];             // 4 KB
  __shared__ __align__(16) float xinLds[64 * 36];          // 9 KB, pitch 36
  __shared__ __align__(16) float w1T[128 * W1T_PITCH];     // 17.4 KB: W1 tile TRANSPOSED [n][k]
  __shared__ float outLds[64 * 2];

  const int e    = blockIdx.y;
  const int m0   = blockIdx.x * 64;
  const int tid  = threadIdx.x;
  const int lane = tid & 31;
  const int wv   = tid >> 5;
  const int wm   = wv & 3;            // row quadrant (16 rows)
  const int wn   = wv >> 2;           // col half (64 cols)
  const int l15  = lane & 15;
  const int khi  = (lane >> 4) << 1;  // 0 or 2 (K offset for upper half-wave)

  for (int i = tid; i < H_SZ; i += 256) impLds[i] = imp[e * H_SZ + i];
  __syncthreads();

  float rowAcc[8];
#pragma unroll
  for (int r = 0; r < 8; ++r) rowAcc[r] = 0.f;

  // xin staging map
  const int xr = tid & 63;          // xin row 0..63
  const int xq = tid >> 6;          // xin k-subblock 0..3
  // W1 staging map (transposed): each thread handles 4 k-pairs for 2 consecutive n
  const int n2 = (tid & 63) * 2;    // 0,2,...,126
  const int kq = tid >> 6;          // 0..3

  for (int n0 = 0; n0 < HID; n0 += 128) {
    v8f acc0 = {0.f,0.f,0.f,0.f,0.f,0.f,0.f,0.f};
    v8f acc1 = acc0, acc2 = acc0, acc3 = acc0;

    for (int kc = 0; kc < KIN; kc += 32) {
      __syncthreads();
      // stage xin[64][32]: cols kc..kc+31 (h*imp | code_emb), built on the fly
      {
        int grow = m0 + xr;
#pragma unroll
        for (int p = 0; p < 2; ++p) {
          int k = kc + xq * 8 + p * 4;
          float4 v;
          if (k < H_SZ) {
            v = *reinterpret_cast<const float4*>(&h[(size_t)grow * H_SZ + k]);
            v.x *= impLds[k + 0]; v.y *= impLds[k + 1];
            v.z *= impLds[k + 2]; v.w *= impLds[k + 3];
          } else {
            v = *reinterpret_cast<const float4*>(&code[(size_t)grow * C_SZ + (k - H_SZ)]);
          }
          *reinterpret_cast<float4*>(&xinLds[xr * 36 + xq * 8 + p * 4]) = v;
        }
      }
      // stage W1 tile TRANSPOSED: w1T[n][k] so each B-fragment is one contiguous b64
      {
#pragma unroll
        for (int i = 0; i < 4; ++i) {
          int kp = kq + 4 * i;                    // k-pair 0..15
          const float* g0 = &W1[((size_t)e * KIN + (kc + 2 * kp)) * HID + n0 + n2];
          float2 r0 = *reinterpret_cast<const float2*>(g0);        // row k,   cols n2,n2+1
          float2 r1 = *reinterpret_cast<const float2*>(g0 + HID);  // row k+1, cols n2,n2+1
          float2 c0 = { r0.x, r1.x };
          float2 c1 = { r0.y, r1.y };
          *reinterpret_cast<float2*>(&w1T[n2 * W1T_PITCH + 2 * kp])       = c0;
          *reinterpret_cast<float2*>(&w1T[(n2 + 1) * W1T_PITCH + 2 * kp]) = c1;
          if (kc + 32 < KIN)
            __builtin_prefetch(g0 + (size_t)32 * HID, 0, 1);
        }
      }
      __syncthreads();

      // 8 K-steps of V_WMMA_F32_16X16X4_F32, 4 C-tiles per wave
#pragma unroll
      for (int ks = 0; ks < 8; ++ks) {
        int kb = ks * 4;
        int ko = kb + khi;
        v2f a = *reinterpret_cast<const v2f*>(&xinLds[(wm * 16 + l15) * 36 + ko]);
        const float* bp = &w1T[(wn * 64 + l15) * W1T_PITCH + ko];
        v2f bf0 = *reinterpret_cast<const v2f*>(bp + 0 * 16 * W1T_PITCH);
        v2f bf1 = *reinterpret_cast<const v2f*>(bp + 1 * 16 * W1T_PITCH);
        v2f bf2 = *reinterpret_cast<const v2f*>(bp + 2 * 16 * W1T_PITCH);
        v2f bf3 = *reinterpret_cast<const v2f*>(bp + 3 * 16 * W1T_PITCH);
        acc0 = __builtin_amdgcn_wmma_f32_16x16x4_f32(false, a, false, bf0, (short)0, acc0, false, false);
        acc1 = __builtin_amdgcn_wmma_f32_16x16x4_f32(false, a, false, bf1, (short)0, acc1, false, false);
        acc2 = __builtin_amdgcn_wmma_f32_16x16x4_f32(false, a, false, bf2, (short)0, acc2, false, false);
        acc3 = __builtin_amdgcn_wmma_f32_16x16x4_f32(false, a, false, bf3, (short)0, acc3, false, false);
      }
    }
    // epilogue: bias + exact GELU + dot with W2[e,:,e]
    int nb = n0 + wn * 64 + l15;
    epilogue_tile(acc0, nb +  0, e, b1, W2, rowAcc);
    epilogue_tile(acc1, nb + 16, e, b1, W2, rowAcc);
    epilogue_tile(acc2, nb + 32, e, b1, W2, rowAcc);
    epilogue_tile(acc3, nb + 48, e, b1, W2, rowAcc);
  }

  // reduce across the 16 lanes sharing each row group
#pragma unroll
  for (int r = 0; r < 8; ++r) {
    float v = rowAcc[r];
    v += __shfl_xor(v, 1, 32);
    v += __shfl_xor(v, 2, 32);
    v += __shfl_xor(v, 4, 32);
    v += __shfl_xor(v, 8, 32);
    rowAcc[r] = v;
  }
  __syncthreads();
  if (lane == 0) {
#pragma unroll
    for (int r = 0; r < 8; ++r) outLds[(wm * 16 + r) * 2 + wn] = rowAcc[r];
  } else if (lane == 16) {
#pragma unroll
    for (int r = 0; r < 8; ++r) outLds[(wm * 16 + 8 + r) * 2 + wn] = rowAcc[r];
  }
  __syncthreads();
  if (tid < 64) {
    int row = m0 + tid;
    float contrib = outLds[tid * 2] + outLds[tid * 2 + 1] + b2[e * E_SZ + e];
    float sw = sortedW[row * E_SZ + e];   // i-th sorted weight for expert i
    atomicAdd(&s[row], contrib * sw);
  }
}

// --------------------------------------------------------- broadcast
__global__ __launch_bounds__(256)
void bcast_kernel(const float* __restrict__ s, float* __restrict__ out) {
  float v = s[blockIdx.x];
  float4 f = { v, v, v, v };
  reinterpret_cast<float4*>(out)[(size_t)blockIdx.x * 256 + threadIdx.x] = f;
}

// --------------------------------------------------------- aux loss
__global__ void aux_kernel(const float* __restrict__ counts, const float* __restrict__ impEnt,
                           float* __restrict__ out) {
  int lane = threadIdx.x;  // block = 32 (one wave)
  float c = counts[lane];
  float tot = wsum32(c);
  float mean = tot * (1.f / 32.f);
  float d = c - mean;
  float var = wsum32(d * d) / 31.f;          // ddof=1
  float sd = sqrtf(var);
  float ld = c / tot;
  float t2 = wsum32(ld * logf(ld + 1e-8f));
  float en = wsum32(impEnt[lane]) * (1.f / 32.f);
  if (lane == 0) out[(size_t)B_SZ * OUTD] = 0.5f * (sd - t2) - 0.01f * en;
}

extern "C" void kernel_launch(void* const* d_in, const int* in_sizes, int n_in,
                              void* d_out, int out_size, void* d_ws, size_t ws_size,
                              hipStream_t stream) {
  const float* h   = (const float*)d_in[0];
  const float* ce  = (const float*)d_in[1];
  const float* gum = (const float*)d_in[2];
  const float* ca  = (const float*)d_in[3];
  const float* fi  = (const float*)d_in[4];
  const float* T   = (const float*)d_in[5];
  const float* W1  = (const float*)d_in[6];
  const float* b1  = (const float*)d_in[7];
  const float* W2  = (const float*)d_in[8];
  const float* b2  = (const float*)d_in[9];
  float* out = (float*)d_out;
  float* ws  = (float*)d_ws;

  float* caN     = ws + WS_CAN;
  float* sortedW = ws + WS_SORTW;
  float* counts  = ws + WS_COUNTS;
  float* imp     = ws + WS_IMP;
  float* impEnt  = ws + WS_IMPENT;
  float* s       = ws + WS_S;

  init_kernel<<<32, 256, 0, stream>>>(s, counts);
  norm_anchor_kernel<<<E_SZ, 32, 0, stream>>>(ca, caN);
  router_kernel<<<B_SZ / 8, 256, 0, stream>>>(ce, gum, caN, sortedW, counts);
  importance_kernel<<<E_SZ, 256, 0, stream>>>(fi, T, imp, impEnt);
  moe_kernel<<<dim3(B_SZ / 64, E_SZ), 256, 0, stream>>>(h, ce, W1, b1, W2, b2, imp, sortedW, s);
  bcast_kernel<<<B_SZ, 256, 0, stream>>>(s, out);
  aux_kernel<<<1, 32, 0, stream>>>(counts, impEnt, out);
}